// VQEmbeddingMovingAverage_9440338116863
// MI455X (gfx1250) — compile-verified
//
#include <hip/hip_runtime.h>

typedef __attribute__((ext_vector_type(2))) float v2f;
typedef __attribute__((ext_vector_type(4))) float v4f;
typedef __attribute__((ext_vector_type(8))) float v8f;

#define NROWS 65536
#define DIMS  256
#define KCODES 1024
#define DECAY 0.99f
#define LDS_STRIDE 260   // 256 + 4 pad -> bank-conflict-free ds_load_b64

// ---------------- K1: zero the scratch accumulators (counts + dw) ----------------
__global__ void vq_zero_kernel(float* __restrict__ p, int n) {
    int i = blockIdx.x * blockDim.x + threadIdx.x;
    if (i < n) p[i] = 0.0f;
}

// ---------------- K2: c_sqr[k] = ||codebook[k]||^2  (one wave per row) ----------
__global__ void vq_csqr_kernel(const float* __restrict__ cb, float* __restrict__ csqr) {
    const int wave = threadIdx.x >> 5;
    const int lane = threadIdx.x & 31;
    const int k = blockIdx.x * 8 + wave;
    const v4f* row = (const v4f*)(cb + (size_t)k * DIMS);
    float s = 0.0f;
#pragma unroll
    for (int j = 0; j < 2; ++j) {
        v4f v = row[j * 32 + lane];
        s += v.x * v.x + v.y * v.y + v.z * v.z + v.w * v.w;
    }
#pragma unroll
    for (int m = 16; m >= 1; m >>= 1) s += __shfl_xor(s, m, 32);
    if (lane == 0) csqr[k] = s;
}

// ---------------- K3: WMMA argmin over the codebook --------------------------
// block = 256 threads (8 waves); each wave owns 16 rows of X; block = 128 rows.
// dist[n,k] = ||c_k||^2 - 2 * x_n . c_k   (x_n^2 is constant per row -> dropped)
__global__ __launch_bounds__(256, 1) void vq_argmin_kernel(
    const float* __restrict__ X, const float* __restrict__ CB,
    const float* __restrict__ csqr, int* __restrict__ indices) {
    __shared__ float cb_s[16 * LDS_STRIDE];

    const int tid  = threadIdx.x;
    const int wave = tid >> 5;
    const int lane = tid & 31;
    const int half = lane >> 4;   // which half-wave: selects K pair {0,1} vs {2,3}
    const int l16  = lane & 15;   // M index for A frags / N index for B frags
    const int row0 = blockIdx.x * 128 + wave * 16;

    // A panel for this wave's 16 rows, resident in VGPRs.
    // 16x4 f32 A layout: lane l<16 -> row l, K=0,1 ; lane l>=16 -> row l-16, K=2,3
    const v2f* xrow = (const v2f*)(X + (size_t)(row0 + l16) * DIMS + 2 * half);
    v2f a[64];
#pragma unroll
    for (int t = 0; t < 64; ++t) a[t] = xrow[2 * t];

    float best[8];
    int   bidx[8];
#pragma unroll
    for (int v = 0; v < 8; ++v) { best[v] = 3.4e38f; bidx[v] = 0; }

    for (int kt = 0; kt < 64; ++kt) {
        const int code0 = kt * 16;
        __syncthreads();
        // cooperative stage of 16 codebook rows into padded LDS
#pragma unroll
        for (int j = 0; j < 4; ++j) {
            int fi = tid + j * 256;            // float4 index 0..1023
            int r  = fi >> 6;                  // 64 float4 per row
            int c4 = fi & 63;
            v4f v = *(const v4f*)(CB + (size_t)(code0 + r) * DIMS + c4 * 4);
            *(v4f*)(cb_s + r * LDS_STRIDE + c4 * 4) = v;
        }
        // prefetch next tile toward L2/WGP while we compute on this one
        if (kt < 63) __builtin_prefetch(CB + (size_t)(code0 + 16) * DIMS + tid * 4, 0, 1);
        __syncthreads();

        v8f acc = {};
        const float* brow = cb_s + l16 * LDS_STRIDE + 2 * half;
#pragma unroll
        for (int t = 0; t < 64; ++t) {
            // B 4x16 f32 layout: lane l<16 -> col l, K=0,1 ; lane l>=16 -> col l-16, K=2,3
            v2f b = *(const v2f*)(brow + 4 * t);
            acc = __builtin_amdgcn_wmma_f32_16x16x4_f32(
                /*neg_a=*/false, a[t], /*neg_b=*/false, b,
                /*c_mod=*/(short)0, acc, /*reuse_a=*/false, /*reuse_b=*/false);
        }

        const int   code = code0 + l16;
        const float cs   = csqr[code];
#pragma unroll
        for (int v = 0; v < 8; ++v) {
            // C layout: VGPR v -> M = v (lanes 0-15) or v+8 (lanes 16-31), N = l16
            float d = cs - 2.0f * acc[v];
            if (d < best[v]) { best[v] = d; bidx[v] = code; }
        }
    }

    // argmin across the 16 lanes of each half (each half holds one row per v)
#pragma unroll
    for (int v = 0; v < 8; ++v) {
        float bv = best[v];
        int   bi = bidx[v];
#pragma unroll
        for (int m = 1; m < 16; m <<= 1) {
            float ov = __shfl_xor(bv, m, 32);
            int   oi = __shfl_xor(bi, m, 32);
            if (ov < bv || (ov == bv && oi < bi)) { bv = ov; bi = oi; }
        }
        if (l16 == 0) indices[row0 + v + 8 * half] = bi;
    }
}

// ---------------- K4: scatter counts + dw via f32 atomics (L2-resident) -------
__global__ void vq_scatter_kernel(const float* __restrict__ Z,
                                  const int* __restrict__ indices,
                                  float* __restrict__ counts,
                                  float* __restrict__ dw) {
    const int wave = threadIdx.x >> 5;
    const int lane = threadIdx.x & 31;
    const int row  = blockIdx.x * 8 + wave;
    const int idx  = indices[row];
    const float* z   = Z  + (size_t)row * DIMS;
    float*       dwr = dw + (size_t)idx * DIMS;
#pragma unroll
    for (int j = 0; j < 2; ++j) {
        int c = (j * 32 + lane) * 4;
        v4f v = *(const v4f*)(z + c);
        atomicAdd(dwr + c + 0, v.x);
        atomicAdd(dwr + c + 1, v.y);
        atomicAdd(dwr + c + 2, v.z);
        atomicAdd(dwr + c + 3, v.w);
    }
    if (lane == 0) atomicAdd(counts + idx, 1.0f);
}

// ---------------- K5: EMA update + new embedding ------------------------------
__global__ void vq_ema_kernel(const float* __restrict__ ema_count,
                              const float* __restrict__ ema_w,
                              const float* __restrict__ counts,
                              const float* __restrict__ dw,
                              float* __restrict__ emb_new,
                              float* __restrict__ cnt_new,
                              float* __restrict__ w_new) {
    const int k = blockIdx.x;
    const int d = threadIdx.x;
    const float cn = DECAY * ema_count[k] + (1.0f - DECAY) * counts[k];
    const size_t o = (size_t)k * DIMS + d;
    const float wn = DECAY * ema_w[o] + (1.0f - DECAY) * dw[o];
    w_new[o]   = wn;
    emb_new[o] = wn / cn;
    if (d == 0) cnt_new[k] = cn;
}

// ---------------- K6: fused double gather z_q_x / z_q_x_bar -------------------
__global__ void vq_gather_kernel(const int* __restrict__ indices,
                                 const float* __restrict__ emb_old,
                                 const float* __restrict__ emb_new,
                                 float* __restrict__ zq,
                                 float* __restrict__ zqb) {
    const int wave = threadIdx.x >> 5;
    const int lane = threadIdx.x & 31;
    const int row  = blockIdx.x * 8 + wave;
    const int idx  = indices[row];
    const v4f* eo = (const v4f*)(emb_old + (size_t)idx * DIMS);
    const v4f* en = (const v4f*)(emb_new + (size_t)idx * DIMS);
    v4f* o1 = (v4f*)(zq  + (size_t)row * DIMS);
    v4f* o2 = (v4f*)(zqb + (size_t)row * DIMS);
#pragma unroll
    for (int j = 0; j < 2; ++j) {
        int c = j * 32 + lane;
        o1[c] = eo[c];
        o2[c] = en[c];
    }
}

extern "C" void kernel_launch(void* const* d_in, const int* in_sizes, int n_in,
                              void* d_out, int out_size, void* d_ws, size_t ws_size,
                              hipStream_t stream) {
    const float* masked_z = (const float*)d_in[0];   // [N, D]
    const float* z_e_x    = (const float*)d_in[1];   // [N, D]
    const float* emb      = (const float*)d_in[2];   // [K, D]
    const float* ema_cnt  = (const float*)d_in[3];   // [K]
    const float* ema_w    = (const float*)d_in[4];   // [K, D]

    // workspace layout (floats): indices[N] | csqr[K] | counts[K] | dw[K*D]
    float* w       = (float*)d_ws;
    int*   indices = (int*)d_ws;
    float* csqr    = w + NROWS;
    float* counts  = w + NROWS + KCODES;
    float* dw      = w + NROWS + 2 * KCODES;

    // output layout (floats): z_q_x | z_q_x_bar | embedding_new | ema_count_new | ema_w_new
    float* out      = (float*)d_out;
    float* zq       = out;
    float* zqb      = out + (size_t)NROWS * DIMS;
    float* emb_new  = out + 2 * (size_t)NROWS * DIMS;
    float* cnt_new  = emb_new + (size_t)KCODES * DIMS;
    float* w_new    = cnt_new + KCODES;

    const int nzero = KCODES + KCODES * DIMS;       // counts + dw are contiguous
    vq_zero_kernel<<<(nzero + 255) / 256, 256, 0, stream>>>(counts, nzero);
    vq_csqr_kernel<<<KCODES / 8, 256, 0, stream>>>(emb, csqr);
    vq_argmin_kernel<<<NROWS / 128, 256, 0, stream>>>(masked_z, emb, csqr, indices);
    vq_scatter_kernel<<<NROWS / 8, 256, 0, stream>>>(z_e_x, indices, counts, dw);
    vq_ema_kernel<<<KCODES, DIMS, 0, stream>>>(ema_cnt, ema_w, counts, dw,
                                               emb_new, cnt_new, w_new);
    vq_gather_kernel<<<NROWS / 8, 256, 0, stream>>>(indices, emb, emb_new, zq, zqb);
}